// SpectralAttention_83116207112859
// MI455X (gfx1250) — compile-verified
//
#include <hip/hip_runtime.h>

#define NB 12
#define D  64
#define NH 2
#define HD 32

typedef __attribute__((ext_vector_type(2))) float v2f;
typedef __attribute__((ext_vector_type(8))) float v8f;

// workspace layout (float offsets)
#define OFF_QW   0              // qw[64], kw[64], vw[64]
#define OFF_QC   192            // qc[12][64], kc[12][64], vc[12][64]
#define OFF_CONST 2496
// const block layout (relative):
#define CN_A 0                  // A[2]   (scaled)
#define CN_P 2                  // P[2]
#define CN_U 4                  // u[2][12] (scaled)
#define CN_V 28                 // v[2][12] (scaled)
#define CN_R 52                 // r[2][12]
#define CN_C 76                 // C[2][16][16] (scaled, padded stride 16)
#define CONST_SZ 588

// ---------------- setup A: projection collapse --------------------
__global__ void sa_setup_a(const float* __restrict__ we, const float* __restrict__ be,
                           const float* __restrict__ bandpos,
                           const float* __restrict__ wq, const float* __restrict__ bq,
                           const float* __restrict__ wk, const float* __restrict__ bk,
                           const float* __restrict__ wv, const float* __restrict__ bv,
                           float* __restrict__ ws) {
  int tid = threadIdx.x;
  const float* W[3]  = {wq, wk, wv};
  const float* Bi[3] = {bq, bk, bv};
  // qw/kw/vw : W @ we
  for (int idx = tid; idx < 3 * D; idx += blockDim.x) {
    int mat = idx / D, i = idx % D;
    const float* w = W[mat];
    float s = 0.f;
    for (int j = 0; j < D; ++j) s = fmaf(w[i * D + j], we[j], s);
    ws[OFF_QW + idx] = s;
  }
  // qc/kc/vc : W @ (be + band_pos[b]) + bias
  for (int idx = tid; idx < 3 * NB * D; idx += blockDim.x) {
    int mat = idx / (NB * D), rem = idx % (NB * D);
    int b = rem / D, i = rem % D;
    const float* w = W[mat];
    float s = Bi[mat][i];
    for (int j = 0; j < D; ++j) s = fmaf(w[i * D + j], be[j] + bandpos[b * D + j], s);
    ws[OFF_QC + idx] = s;
  }
}

// ---------------- setup B: WMMA Gram matrices + small dots --------
// launched with exactly 1 wave (32 threads): EXEC all-ones for WMMA
__global__ void sa_setup_b(const float* __restrict__ wo, float* __restrict__ ws) {
  const int lane = threadIdx.x;          // 0..31
  const int m    = lane & 15;            // row (A) / col (B,D)
  const int hi   = lane >> 4;            // K-half selector
  const float* qw = ws + OFF_QW;
  const float* kw = ws + OFF_QW + D;
  const float* vw = ws + OFF_QW + 2 * D;
  const float* qc = ws + OFF_QC;               // [12][64] (rows 12..15 read harmless garbage)
  const float* kc = ws + OFF_QC + NB * D;
  const float* vc = ws + OFF_QC + 2 * NB * D;
  float* cst = ws + OFF_CONST;
  const float scale = 0.17677669529663687f;    // 1/sqrt(32)

  // C_h = qc_h [16x32] @ kc_h^T [32x16], exact f32 via V_WMMA_F32_16X16X4_F32 chain
  for (int h = 0; h < NH; ++h) {
    v8f acc = {};
    for (int k0 = 0; k0 < HD; k0 += 4) {
      int c = h * HD + k0 + 2 * hi;
      v2f a, b;
      a.x = qc[m * D + c];     a.y = qc[m * D + c + 1];   // A[m][k]
      b.x = kc[m * D + c];     b.y = kc[m * D + c + 1];   // B[k][n] = kc[n][k]
      acc = __builtin_amdgcn_wmma_f32_16x16x4_f32(false, a, false, b,
                                                  (short)0, acc, false, false);
    }
    #pragma unroll
    for (int r = 0; r < 8; ++r) {
      int row = r + 8 * hi;                                // D row
      cst[CN_C + h * 256 + row * 16 + m] = acc[r] * scale; // D[row][m]
    }
  }

  // A_h = scale * qw_h.kw_h ; P_h = vw_h.wo_h
  if (lane < NH) {
    int h = lane;
    float a = 0.f, p = 0.f;
    for (int i = 0; i < HD; ++i) {
      a = fmaf(qw[h * HD + i], kw[h * HD + i], a);
      p = fmaf(vw[h * HD + i], wo[h * HD + i], p);
    }
    cst[CN_A + h] = a * scale;
    cst[CN_P + h] = p;
  }

  // u_h[b] = qw_h.kc_h[b] ; v_h[b] = qc_h[b].kw_h ; r_h[b] = vc_h[b].wo_h
  for (int idx = lane; idx < 72; idx += 32) {
    int kind = idx / 24, rem = idx % 24, h = rem / 12, b = rem % 12;
    float s = 0.f;
    if (kind == 0) {
      for (int i = 0; i < HD; ++i) s = fmaf(qw[h * HD + i], kc[b * D + h * HD + i], s);
      cst[CN_U + h * NB + b] = s * scale;
    } else if (kind == 1) {
      for (int i = 0; i < HD; ++i) s = fmaf(qc[b * D + h * HD + i], kw[h * HD + i], s);
      cst[CN_V + h * NB + b] = s * scale;
    } else {
      for (int i = 0; i < HD; ++i) s = fmaf(vc[b * D + h * HD + i], wo[h * HD + i], s);
      cst[CN_R + h * NB + b] = s;
    }
  }
}

// ---------------- main: 1 thread per token, memory-bound ----------
__global__ void __launch_bounds__(256)
sa_main(const float* __restrict__ x, const unsigned char* __restrict__ mask,
        const float* __restrict__ ws, const float* __restrict__ bo_p,
        float* __restrict__ out, int ntok) {
  __shared__ float cst[CONST_SZ];
  for (int i = threadIdx.x; i < CONST_SZ; i += blockDim.x) cst[i] = ws[OFF_CONST + i];
  __syncthreads();

  int n = blockIdx.x * blockDim.x + threadIdx.x;
  if (n >= ntok) return;
  const float bo = bo_p[0];

  // load x[n, 0..11] (48B, 16B aligned) and mask bytes (12B, 4B aligned)
  const float4* xp = (const float4*)(x + (size_t)n * NB);
  float4 x0 = xp[0], x1 = xp[1], x2 = xp[2];
  float xv[NB] = {x0.x, x0.y, x0.z, x0.w, x1.x, x1.y, x1.z, x1.w,
                  x2.x, x2.y, x2.z, x2.w};
  const unsigned int* mp = (const unsigned int*)(mask + (size_t)n * NB);
  unsigned int mA = mp[0], mB = mp[1], mC = mp[2];
  bool mb[NB];
  mb[0] = (mA & 0xffu) != 0;  mb[1] = (mA >> 8 & 0xffu) != 0;
  mb[2] = (mA >> 16 & 0xffu) != 0; mb[3] = (mA >> 24) != 0;
  mb[4] = (mB & 0xffu) != 0;  mb[5] = (mB >> 8 & 0xffu) != 0;
  mb[6] = (mB >> 16 & 0xffu) != 0; mb[7] = (mB >> 24) != 0;
  mb[8] = (mC & 0xffu) != 0;  mb[9] = (mC >> 8 & 0xffu) != 0;
  mb[10] = (mC >> 16 & 0xffu) != 0; mb[11] = (mC >> 24) != 0;

  float delta[NB];
  #pragma unroll
  for (int b = 0; b < NB; ++b) delta[b] = 0.f;

  #pragma unroll
  for (int h = 0; h < NH; ++h) {
    const float Ah = cst[CN_A + h];
    const float Ph = cst[CN_P + h];
    const float* uh = &cst[CN_U + h * NB];
    const float* vh = &cst[CN_V + h * NB];
    const float* rh = &cst[CN_R + h * NB];
    const float* Ch = &cst[CN_C + h * 256];
    float g[NB], w[NB];
    #pragma unroll
    for (int j = 0; j < NB; ++j) {
      g[j] = fmaf(xv[j], Ah, uh[j]);   // scale folded into Ah,uh
      w[j] = fmaf(xv[j], Ph, rh[j]);   // per-key value·wo
    }
    #pragma unroll
    for (int b = 0; b < NB; ++b) {
      float vb = vh[b];
      float s[NB];
      float rmax = -3.0e38f;
      #pragma unroll
      for (int j = 0; j < NB; ++j) {
        float t = fmaf(xv[b], g[j], fmaf(xv[j], vb, Ch[b * 16 + j]));
        s[j] = t;
        if (mb[j] && t > rmax) rmax = t;
      }
      float num = 0.f, den = 0.f;
      #pragma unroll
      for (int j = 0; j < NB; ++j) {
        float e = mb[j] ? __expf(s[j] - rmax) : 0.f;
        den += e;
        num = fmaf(e, w[j], num);
      }
      if (den > 0.f) delta[b] += num / den;   // all-masked row -> 0 (nan_to_num semantics)
    }
  }

  float4 o0, o1, o2;
  float y[NB];
  #pragma unroll
  for (int b = 0; b < NB; ++b)
    y[b] = xv[b] + (mb[b] ? (delta[b] + bo) : 0.f);
  o0.x = y[0]; o0.y = y[1]; o0.z = y[2];  o0.w = y[3];
  o1.x = y[4]; o1.y = y[5]; o1.z = y[6];  o1.w = y[7];
  o2.x = y[8]; o2.y = y[9]; o2.z = y[10]; o2.w = y[11];
  float4* op = (float4*)(out + (size_t)n * NB);
  op[0] = o0; op[1] = o1; op[2] = o2;
}

extern "C" void kernel_launch(void* const* d_in, const int* in_sizes, int n_in,
                              void* d_out, int out_size, void* d_ws, size_t ws_size,
                              hipStream_t stream) {
  (void)n_in; (void)out_size; (void)ws_size;
  const float* x          = (const float*)d_in[0];
  const unsigned char* bm = (const unsigned char*)d_in[1];
  const float* we  = (const float*)d_in[2];
  const float* be  = (const float*)d_in[3];
  const float* bp  = (const float*)d_in[4];
  const float* wq  = (const float*)d_in[5];
  const float* bq  = (const float*)d_in[6];
  const float* wk  = (const float*)d_in[7];
  const float* bk  = (const float*)d_in[8];
  const float* wv  = (const float*)d_in[9];
  const float* bv  = (const float*)d_in[10];
  const float* wo  = (const float*)d_in[11];
  const float* bo  = (const float*)d_in[12];
  float* ws  = (float*)d_ws;
  float* out = (float*)d_out;
  int ntok = in_sizes[0] / NB;

  sa_setup_a<<<1, 256, 0, stream>>>(we, be, bp, wq, bq, wk, bk, wv, bv, ws);
  sa_setup_b<<<1, 32, 0, stream>>>(wo, ws);
  sa_main<<<(ntok + 255) / 256, 256, 0, stream>>>(x, bm, ws, bo, out, ntok);
}